// KimiLinearKDA_42803644072883
// MI455X (gfx1250) — compile-verified
//
#include <hip/hip_runtime.h>
#include <hip/hip_bf16.h>

// ---------------------------------------------------------------------------
// KDA prefill for MI455X (gfx1250): chunked gated delta rule with f32 WMMA.
//   T=4096, H=16 heads, DK=DV=128, conv K=4, chunk C=64.
// Kernel 1: depthwise causal conv1d + SiLU + split + l2norm + gate/beta.
// Kernel 2: per-head chunk scan, one workgroup (8 wave32) per head,
//           state S[128x128] resident in LDS, matmuls via v_wmma_f32_16x16x4.
//           g/v chunk staging uses gfx1250 async global->LDS copies
//           (global_load_async_to_lds_b128 + s_wait_asynccnt).
// ---------------------------------------------------------------------------

#define T_TOK 4096
#define NH 16
#define DK 128
#define DV 128
#define KW 4
#define QKV_C 6144
#define CHUNK 64
#define NCHUNK (T_TOK / CHUNK)

typedef __attribute__((ext_vector_type(2))) float v2f;
typedef __attribute__((ext_vector_type(8))) float v8f;

__device__ __forceinline__ v8f wmma4(v2f a, v2f b, v8f c) {
  // D = A(16x4,f32) * B(4x16,f32) + C(16x16,f32)
  return __builtin_amdgcn_wmma_f32_16x16x4_f32(
      /*neg_a=*/false, a, /*neg_b=*/false, b,
      /*c_mod=*/(short)0, c, /*reuse_a=*/false, /*reuse_b=*/false);
}

// gfx1250 async global->LDS copy, 16B per lane. lds_off is the byte offset
// into the workgroup's LDS block (dynamic LDS base == 0: no static LDS here).
__device__ __forceinline__ void async_g2l_b128(unsigned lds_off, const void* gaddr) {
  asm volatile("global_load_async_to_lds_b128 %0, %1, off"
               :: "v"(lds_off), "v"(gaddr)
               : "memory");
}
__device__ __forceinline__ void wait_async0() {
  asm volatile("s_wait_asynccnt 0x0" ::: "memory");
}

// ------------------------- Kernel 1: preprocessing -------------------------
__global__ __launch_bounds__(256) void kda_preprocess_kernel(
    const float* __restrict__ x,        // [T, 6144] mixed_qkv
    const float* __restrict__ fg,       // [T, 2048] forget_gate
    const float* __restrict__ beta,     // [T, 16]
    const float* __restrict__ w,        // [6144, 4] conv weights
    const float* __restrict__ A_log,    // [16]
    const float* __restrict__ dt_bias,  // [16, 128]
    float* __restrict__ q_ws,           // [H][T][DK]
    float* __restrict__ k_ws,           // [H][T][DK]
    float* __restrict__ v_ws,           // [H][T][DV]
    float* __restrict__ g_ws,           // [H][T][DK]
    float* __restrict__ b_ws)           // [H][T]
{
  const int t = blockIdx.x;
  const int tid = threadIdx.x;
  __shared__ float ybuf[QKV_C];
  __shared__ float rnorm[32];

  // causal depthwise conv + SiLU
  for (int c = tid; c < QKV_C; c += 256) {
    float acc = 0.f;
#pragma unroll
    for (int j = 0; j < KW; ++j) {
      const int tt = t - (KW - 1) + j;
      const float xv = (tt >= 0) ? x[tt * QKV_C + c] : 0.f;
      acc += xv * w[c * KW + j];
    }
    acc = acc / (1.f + __expf(-acc));  // SiLU
    ybuf[c] = acc;
  }
  __syncthreads();

  // l2 norms: 32 groups of 128 (q heads then k heads)
  const int wv = tid >> 5, ln = tid & 31;
  for (int grp = wv; grp < 32; grp += 8) {
    const float* base = ybuf + grp * DK;
    float s = 0.f;
    for (int i = ln; i < DK; i += 32) { const float v = base[i]; s += v * v; }
#pragma unroll
    for (int off = 16; off; off >>= 1) s += __shfl_xor(s, off, 32);
    if (ln == 0) rnorm[grp] = rsqrtf(s + 1e-6f);
  }
  __syncthreads();

  const float qscale = 0.08838834764831845f;  // 128^-0.5
  for (int c = tid; c < NH * DK; c += 256) {
    const int h = c >> 7, d = c & 127;
    const long hb = (long)h * T_TOK * DK + (long)t * DK + d;
    q_ws[hb] = ybuf[c] * rnorm[h] * qscale;
    k_ws[hb] = ybuf[2048 + c] * rnorm[16 + h];
    v_ws[hb] = ybuf[4096 + c];
    // fused KDA gate: g = -exp(A_log) * softplus(fg + dt_bias)
    const float gv = fg[(long)t * 2048 + c] + dt_bias[c];
    const float sp = (gv > 20.f) ? gv : log1pf(__expf(gv));
    g_ws[hb] = -__expf(A_log[h]) * sp;
  }
  if (tid < NH) {
    const float bv = beta[t * NH + tid];
    b_ws[(long)tid * T_TOK + t] = 1.f / (1.f + __expf(-bv));
  }
}

// ------------------------- Kernel 2: chunked scan --------------------------
// LDS layout (floats), dynamic shared memory:
#define LDS_K 132  // row stride for [*,128] buffers (bank-conflict pad)
#define LDA 66     // row stride for [64,64] buffers
#define KG_OFF 0
#define KD_OFF (KG_OFF + CHUNK * LDS_K)
#define QG_OFF (KD_OFF + CHUNK * LDS_K)
#define S_OFF (QG_OFF + CHUNK * LDS_K)
#define U_OFF (S_OFF + DK * LDS_K)  // doubles as g-buffer then v/rhs/u buffer
#define A_OFF (U_OFF + CHUNK * LDS_K)
#define AQ_OFF (A_OFF + CHUNK * LDA)
#define B_OFF (AQ_OFF + CHUNK * LDA)
#define EGC_OFF (B_OFF + CHUNK)
#define SMEM_FLOATS (EGC_OFF + DK)
#define SMEM_BYTES (SMEM_FLOATS * 4)

__global__ __launch_bounds__(256, 1) void kda_chunk_scan_kernel(
    const float* __restrict__ q_ws, const float* __restrict__ k_ws,
    const float* __restrict__ v_ws, const float* __restrict__ g_ws,
    const float* __restrict__ b_ws, float* __restrict__ out)  // [T][H][DV]
{
  extern __shared__ float sm[];
  float* kg = sm + KG_OFF;
  float* kd = sm + KD_OFF;
  float* qg = sm + QG_OFF;
  float* Sb = sm + S_OFF;
  float* gb = sm + U_OFF;  // g / cumsum G
  float* ub = sm + U_OFF;  // later: v -> rhs -> u (same space, after barrier)
  float* Ab = sm + A_OFF;
  float* Aqb = sm + AQ_OFF;
  float* bb = sm + B_OFF;
  float* egc = sm + EGC_OFF;

  const int h = blockIdx.x;
  const int tid = threadIdx.x;
  const int ln = tid & 31, wv = tid >> 5;
  const int half = ln >> 4, l = ln & 15;

  const float* qh = q_ws + (long)h * T_TOK * DK;
  const float* kh = k_ws + (long)h * T_TOK * DK;
  const float* vh = v_ws + (long)h * T_TOK * DV;
  const float* gh = g_ws + (long)h * T_TOK * DK;
  const float* bh = b_ws + (long)h * T_TOK;

  // Tile ownership (all phases): one row-tile per wave-group, multiple
  // col-tiles per wave -> shared A-fragment, independent WMMA chains.
  const int tr4 = wv >> 1;            // 0..3  (64-row matrices: rhs/o)
  const int tc4 = (wv & 1) * 4;       // col-tile base (x4) for 128-col outputs
  const int tc2 = (wv & 1) * 2;       // col-tile base (x2) for 64-col outputs

  // zero the state
  for (int i = tid; i < DK * LDS_K; i += 256) Sb[i] = 0.f;
  __syncthreads();

  for (int cc = 0; cc < NCHUNK; ++cc) {
    const int t0 = cc * CHUNK;

    // ---- P1: async-load g chunk to LDS; load b; prefetch next chunk ----
    for (int i = tid; i < CHUNK * DK / 4; i += 256) {
      const int e = i * 4, r = e >> 7, d = e & 127;
      async_g2l_b128((unsigned)((U_OFF + r * LDS_K + d) * 4),
                     &gh[(long)(t0 + r) * DK + d]);
    }
    if (tid < CHUNK) bb[tid] = bh[t0 + tid];
    if (cc + 1 < NCHUNK) {
      const int r = tid >> 2, d = (tid & 3) * 32;
      const long off = (long)((cc + 1) * CHUNK + r) * DK + d;
      __builtin_prefetch(&qh[off], 0, 1);
      __builtin_prefetch(&kh[off], 0, 1);
      __builtin_prefetch(&vh[off], 0, 1);
      __builtin_prefetch(&gh[off], 0, 1);
    }
    wait_async0();
    __syncthreads();

    // ---- P2: within-chunk inclusive cumsum of g (per DK column) ----
    if (tid < DK) {
      float s = 0.f;
      for (int r = 0; r < CHUNK; ++r) {
        s += gb[r * LDS_K + tid];
        gb[r * LDS_K + tid] = s;
      }
      egc[tid] = __expf(s);  // chunk-end decay exp(G_C)
    }
    __syncthreads();

    // ---- P3: kg = k*e^G, kd = k*e^-G, qg = q*e^G ----
    for (int i = tid; i < CHUNK * DK; i += 256) {
      const int r = i >> 7, d = i & 127;
      const float G = gb[r * LDS_K + d];
      const float eg = __expf(G), em = __expf(-G);
      const float kv = kh[(long)(t0 + r) * DK + d];
      kg[r * LDS_K + d] = kv * eg;
      kd[r * LDS_K + d] = kv * em;
      qg[r * LDS_K + d] = qh[(long)(t0 + r) * DK + d] * eg;
    }
    __syncthreads();

    // ---- P4a: async-load v chunk into u-buffer (overwrites g space) ----
    for (int i = tid; i < CHUNK * DV / 4; i += 256) {
      const int e = i * 4, r = e >> 7, d = e & 127;
      async_g2l_b128((unsigned)((U_OFF + r * LDS_K + d) * 4),
                     &vh[(long)(t0 + r) * DV + d]);
    }
    wait_async0();
    __syncthreads();

    // ---- P4b: WMMA matmul set 1 ----
    {
      // A = kg @ kd^T  (64x64, K=128): row-tile tr4, 2 col-tiles, 2 chains
      const int r0 = tr4 * 16;
      v8f acc0 = {}, acc1 = {};
      for (int kk = 0; kk < DK; kk += 4) {
        v2f a, b0, b1;
        a.x = kg[(r0 + l) * LDS_K + kk + 2 * half + 0];
        a.y = kg[(r0 + l) * LDS_K + kk + 2 * half + 1];
        b0.x = kd[(tc2 * 16 + l) * LDS_K + kk + 2 * half + 0];  // kd^T
        b0.y = kd[(tc2 * 16 + l) * LDS_K + kk + 2 * half + 1];
        b1.x = kd[((tc2 + 1) * 16 + l) * LDS_K + kk + 2 * half + 0];
        b1.y = kd[((tc2 + 1) * 16 + l) * LDS_K + kk + 2 * half + 1];
        acc0 = wmma4(a, b0, acc0);
        acc1 = wmma4(a, b1, acc1);
      }
#pragma unroll
      for (int i = 0; i < 8; ++i) {
        Ab[(r0 + i + 8 * half) * LDA + tc2 * 16 + l] = acc0[i];
        Ab[(r0 + i + 8 * half) * LDA + (tc2 + 1) * 16 + l] = acc1[i];
      }
    }
    {
      // Aq = qg @ kd^T  (64x64, K=128): 2 chains
      const int r0 = tr4 * 16;
      v8f acc0 = {}, acc1 = {};
      for (int kk = 0; kk < DK; kk += 4) {
        v2f a, b0, b1;
        a.x = qg[(r0 + l) * LDS_K + kk + 2 * half + 0];
        a.y = qg[(r0 + l) * LDS_K + kk + 2 * half + 1];
        b0.x = kd[(tc2 * 16 + l) * LDS_K + kk + 2 * half + 0];
        b0.y = kd[(tc2 * 16 + l) * LDS_K + kk + 2 * half + 1];
        b1.x = kd[((tc2 + 1) * 16 + l) * LDS_K + kk + 2 * half + 0];
        b1.y = kd[((tc2 + 1) * 16 + l) * LDS_K + kk + 2 * half + 1];
        acc0 = wmma4(a, b0, acc0);
        acc1 = wmma4(a, b1, acc1);
      }
#pragma unroll
      for (int i = 0; i < 8; ++i) {
        Aqb[(r0 + i + 8 * half) * LDA + tc2 * 16 + l] = acc0[i];
        Aqb[(r0 + i + 8 * half) * LDA + (tc2 + 1) * 16 + l] = acc1[i];
      }
    }
    {
      // rhs = b * (v - kg @ S0)  (64x128, K=128): 4 chains, shared A-frag
      const int r0 = tr4 * 16;
      v8f acc[4] = {};
      for (int kk = 0; kk < DK; kk += 4) {
        v2f a;
        a.x = kg[(r0 + l) * LDS_K + kk + 2 * half + 0];
        a.y = kg[(r0 + l) * LDS_K + kk + 2 * half + 1];
#pragma unroll
        for (int j = 0; j < 4; ++j) {
          const int c0 = (tc4 + j) * 16;
          v2f b;
          b.x = Sb[(kk + 2 * half + 0) * LDS_K + c0 + l];
          b.y = Sb[(kk + 2 * half + 1) * LDS_K + c0 + l];
          acc[j] = wmma4(a, b, acc[j]);
        }
      }
#pragma unroll
      for (int j = 0; j < 4; ++j) {
        const int c0 = (tc4 + j) * 16;
#pragma unroll
        for (int i = 0; i < 8; ++i) {
          const int rr = r0 + i + 8 * half, dd = c0 + l;
          ub[rr * LDS_K + dd] = bb[rr] * (ub[rr * LDS_K + dd] - acc[j][i]);
        }
      }
    }
    __syncthreads();

    // ---- P5: mask Aq to lower-triangular (inclusive diagonal) ----
    for (int i = tid; i < CHUNK * CHUNK; i += 256) {
      const int r = i >> 6, c = i & 63;
      if (c > r) Aqb[r * LDA + c] = 0.f;
    }
    __syncthreads();

    // ---- P6: forward substitution  u = (I + diag(b) A_strict)^-1 rhs ----
    for (int i = 1; i < CHUNK; ++i) {
      if (tid < DV) {
        float s = 0.f;
        for (int j = 0; j < i; ++j) s += Ab[i * LDA + j] * ub[j * LDS_K + tid];
        ub[i * LDS_K + tid] -= bb[i] * s;
      }
      __syncthreads();
    }

    // ---- P7: o = qg @ S0 + tril(Aq) @ u ; single fused accumulator chain ----
    {
      const int r0 = tr4 * 16;
      v8f acc[4] = {};
      for (int kk = 0; kk < DK; kk += 4) {  // qg @ S0
        v2f a;
        a.x = qg[(r0 + l) * LDS_K + kk + 2 * half + 0];
        a.y = qg[(r0 + l) * LDS_K + kk + 2 * half + 1];
#pragma unroll
        for (int j = 0; j < 4; ++j) {
          const int c0 = (tc4 + j) * 16;
          v2f b;
          b.x = Sb[(kk + 2 * half + 0) * LDS_K + c0 + l];
          b.y = Sb[(kk + 2 * half + 1) * LDS_K + c0 + l];
          acc[j] = wmma4(a, b, acc[j]);
        }
      }
      for (int kk = 0; kk < CHUNK; kk += 4) {  // + tril(Aq) @ u
        v2f a;
        a.x = Aqb[(r0 + l) * LDA + kk + 2 * half + 0];
        a.y = Aqb[(r0 + l) * LDA + kk + 2 * half + 1];
#pragma unroll
        for (int j = 0; j < 4; ++j) {
          const int c0 = (tc4 + j) * 16;
          v2f b;
          b.x = ub[(kk + 2 * half + 0) * LDS_K + c0 + l];
          b.y = ub[(kk + 2 * half + 1) * LDS_K + c0 + l];
          acc[j] = wmma4(a, b, acc[j]);
        }
      }
#pragma unroll
      for (int j = 0; j < 4; ++j) {
        const int c0 = (tc4 + j) * 16;
#pragma unroll
        for (int i = 0; i < 8; ++i) {
          const int rr = r0 + i + 8 * half;
          out[(long)(t0 + rr) * (NH * DV) + h * DV + c0 + l] = acc[j][i];
        }
      }
    }

    // ---- P8: state update  S = e^{G_C} * (S + kd^T @ u) ----
    // Wave owns S row-tile wv (16 rows), all 8 col-tiles, 2 batches of 4.
    {
      const int m0 = wv * 16;
#pragma unroll
      for (int batch = 0; batch < 2; ++batch) {
        const int cb = batch * 4;
        v8f acc[4];
#pragma unroll
        for (int j = 0; j < 4; ++j) {
          const int c0 = (cb + j) * 16;
#pragma unroll
          for (int i = 0; i < 8; ++i)
            acc[j][i] = Sb[(m0 + i + 8 * half) * LDS_K + c0 + l];
        }
        for (int kk = 0; kk < CHUNK; kk += 4) {
          v2f a;  // kd^T (DK x C): shared across the 4 col-tiles
          a.x = kd[(kk + 2 * half + 0) * LDS_K + m0 + l];
          a.y = kd[(kk + 2 * half + 1) * LDS_K + m0 + l];
#pragma unroll
          for (int j = 0; j < 4; ++j) {
            const int c0 = (cb + j) * 16;
            v2f b;
            b.x = ub[(kk + 2 * half + 0) * LDS_K + c0 + l];
            b.y = ub[(kk + 2 * half + 1) * LDS_K + c0 + l];
            acc[j] = wmma4(a, b, acc[j]);
          }
        }
#pragma unroll
        for (int j = 0; j < 4; ++j) {
          const int c0 = (cb + j) * 16;
#pragma unroll
          for (int i = 0; i < 8; ++i) {
            const int mm = m0 + i + 8 * half;
            Sb[mm * LDS_K + c0 + l] = egc[mm] * acc[j][i];
          }
        }
      }
    }
    __syncthreads();
  }
}

// ------------------------------ launcher -----------------------------------
extern "C" void kernel_launch(void* const* d_in, const int* in_sizes, int n_in,
                              void* d_out, int out_size, void* d_ws, size_t ws_size,
                              hipStream_t stream) {
  const float* mixed_qkv = (const float*)d_in[0];
  const float* forget_gate = (const float*)d_in[1];
  const float* beta = (const float*)d_in[2];
  const float* conv_w = (const float*)d_in[3];
  const float* A_log = (const float*)d_in[4];
  const float* dt_bias = (const float*)d_in[5];
  float* out = (float*)d_out;

  const long HTD = (long)NH * T_TOK * DK;  // 8,388,608 floats
  float* ws = (float*)d_ws;
  float* q_ws = ws;
  float* k_ws = q_ws + HTD;
  float* v_ws = k_ws + HTD;
  float* g_ws = v_ws + HTD;
  float* b_ws = g_ws + HTD;

  kda_preprocess_kernel<<<T_TOK, 256, 0, stream>>>(
      mixed_qkv, forget_gate, beta, conv_w, A_log, dt_bias,
      q_ws, k_ws, v_ws, g_ws, b_ws);

  hipFuncSetAttribute((const void*)kda_chunk_scan_kernel,
                      hipFuncAttributeMaxDynamicSharedMemorySize, SMEM_BYTES);
  kda_chunk_scan_kernel<<<NH, 256, SMEM_BYTES, stream>>>(
      q_ws, k_ws, v_ws, g_ws, b_ws, out);
}